// GraphAttentionConvBinaryClassifier_74947179315799
// MI455X (gfx1250) — compile-verified
//
#include <hip/hip_runtime.h>
#include <hip/hip_bf16.h>

typedef float v2f __attribute__((ext_vector_type(2)));
typedef float v8f __attribute__((ext_vector_type(8)));

#define D_H 128
#define NUM_GRAPHS 64
#define NEG_SLOPE 0.2f
#define SX_STRIDE 132   // 16-row X slab padding (132 % 64 = 4 -> conflict-free A reads)
#define SW_STRIDE 144   // W panel padding (144 % 64 = 16 -> half-waves hit disjoint banks)

__device__ __forceinline__ float leaky(float v) { return v > 0.f ? v : NEG_SLOPE * v; }

// order-preserving float->uint encoding for atomicMax-based float max
__device__ __forceinline__ unsigned enc_ord(float f) {
    unsigned u = __float_as_uint(f);
    return (f >= 0.f) ? (u | 0x80000000u) : ~u;
}
__device__ __forceinline__ float dec_ord(unsigned u) {
    return (u & 0x80000000u) ? __uint_as_float(u & 0x7FFFFFFFu) : __uint_as_float(~u);
}

// ---------------------------------------------------------------------------
// feat = X @ W   (N x 128) * (128 x 128), fp32 WMMA 16x16x4.
// Block = 256 threads = 8 waves; block computes 16 rows x 128 cols;
// wave w owns the 16x16 tile at columns [16w, 16w+16).
// W is staged through LDS in two 64-row K-stages; X slab staged once;
// output goes back through LDS for coalesced b128 stores.
// ---------------------------------------------------------------------------
__global__ void __launch_bounds__(256) gat_gemm_wmma(
    const float* __restrict__ X, const float* __restrict__ W,
    float* __restrict__ feat, int N)
{
    __shared__ float sX[16 * SX_STRIDE];    //  8.4 KB (A slab, reused for output tile)
    __shared__ float sW[64 * SW_STRIDE];    // 36.9 KB (one K-stage of W)

    const int tid  = threadIdx.x;
    const int wave = tid >> 5;
    const int lane = tid & 31;
    const int row0 = blockIdx.x * 16;

    // stage the 16x128 X slab into LDS (coalesced b128 reads)
    for (int i = tid; i < 512; i += 256) {
        int r  = i >> 5;
        int c4 = (i & 31) << 2;
        int gr = row0 + r; if (gr >= N) gr = N - 1;
        const float4 v = *(const float4*)(X + (size_t)gr * D_H + c4);
        float* p = sX + r * SX_STRIDE + c4;
        p[0] = v.x; p[1] = v.y; p[2] = v.z; p[3] = v.w;
    }

    const int col0  = wave << 4;
    const int idx16 = lane & 15;     // A row / B col for this lane
    const int half  = lane >> 4;     // K sub-pair selector (0 -> K0,1 ; 1 -> K2,3)

    const float* aRow = sX + idx16 * SX_STRIDE + 2 * half;
    const float* bCol = sW + (2 * half) * SW_STRIDE + col0 + idx16;

    v8f acc = {};
    for (int ks = 0; ks < D_H; ks += 64) {
        // stage 64 rows of W into LDS (coalesced b128 reads)
        __syncthreads();
        for (int i = tid; i < 2048; i += 256) {
            int r  = i >> 5;
            int c4 = (i & 31) << 2;
            const float4 v = *(const float4*)(W + (size_t)(ks + r) * D_H + c4);
            float* p = sW + r * SW_STRIDE + c4;
            p[0] = v.x; p[1] = v.y; p[2] = v.z; p[3] = v.w;
        }
        __syncthreads();

        #pragma unroll
        for (int k = 0; k < 64; k += 4) {
            v2f a, b;
            a.x = aRow[ks + k];
            a.y = aRow[ks + k + 1];
            b.x = bCol[k * SW_STRIDE];
            b.y = bCol[(k + 1) * SW_STRIDE];
            acc = __builtin_amdgcn_wmma_f32_16x16x4_f32(
                      false, a, false, b, (short)0, acc, false, false);
        }
    }

    // deposit the 16x16 tile into the (now dead) X slab: half-waves write
    // 32 banks apart -> conflict-free
    __syncthreads();
    {
        const int colw  = col0 + idx16;
        const int rbase = half * 8;
        #pragma unroll
        for (int r = 0; r < 8; ++r)
            sX[(rbase + r) * SX_STRIDE + colw] = acc[r];
    }
    __syncthreads();

    // coalesced b128 stores of the 16x128 output slab
    for (int i = tid; i < 512; i += 256) {
        int r  = i >> 5;
        int c4 = (i & 31) << 2;
        int gr = row0 + r;
        if (gr < N) {
            const float* p = sX + r * SX_STRIDE + c4;
            float4 v = { p[0], p[1], p[2], p[3] };
            *(float4*)(feat + (size_t)gr * D_H + c4) = v;
        }
    }
}

// ---------------------------------------------------------------------------
// el[n] = feat[n] . a_l ; er[n] = feat[n] . a_r    (one wave per node)
// ---------------------------------------------------------------------------
__global__ void __launch_bounds__(256) node_attn(
    const float* __restrict__ feat, const float* __restrict__ al,
    const float* __restrict__ ar, float* __restrict__ el,
    float* __restrict__ er, int N)
{
    int w    = (blockIdx.x * 256 + threadIdx.x) >> 5;
    int lane = threadIdx.x & 31;
    if (w >= N) return;
    float4 v = ((const float4*)(feat + (size_t)w * D_H))[lane];
    float4 L = ((const float4*)al)[lane];
    float4 R = ((const float4*)ar)[lane];
    float sl = v.x * L.x + v.y * L.y + v.z * L.z + v.w * L.w;
    float sr = v.x * R.x + v.y * R.y + v.z * R.z + v.w * R.w;
    #pragma unroll
    for (int off = 16; off > 0; off >>= 1) {
        sl += __shfl_xor(sl, off, 32);
        sr += __shfl_xor(sr, off, 32);
    }
    if (lane == 0) { el[w] = sl; er[w] = sr; }
}

// ---------------------------------------------------------------------------
// pass 1: per-dst max of e (ordered-uint atomicMax)
// ---------------------------------------------------------------------------
__global__ void __launch_bounds__(256) edge_max(
    const int* __restrict__ src, const int* __restrict__ dst,
    const float* __restrict__ el, const float* __restrict__ er,
    unsigned* __restrict__ menc, int E)
{
    int e = blockIdx.x * 256 + threadIdx.x;
    if (e >= E) return;
    int dn = dst[e];
    float v = leaky(el[src[e]] + er[dn]);
    atomicMax(&menc[dn], enc_ord(v));
}

// ---------------------------------------------------------------------------
// pass 2: s[dst] += exp(e - m[dst])
// ---------------------------------------------------------------------------
__global__ void __launch_bounds__(256) edge_sum(
    const int* __restrict__ src, const int* __restrict__ dst,
    const float* __restrict__ el, const float* __restrict__ er,
    const unsigned* __restrict__ menc, float* __restrict__ s, int E)
{
    int e = blockIdx.x * 256 + threadIdx.x;
    if (e >= E) return;
    int dn = dst[e];
    float v = leaky(el[src[e]] + er[dn]);
    atomicAdd(&s[dn], expf(v - dec_ord(menc[dn])));
}

// ---------------------------------------------------------------------------
// pass 3: out[dst] += alpha * feat[src]   (one wave per edge, 4 feats/lane)
// ---------------------------------------------------------------------------
__global__ void __launch_bounds__(256) edge_aggr(
    const int* __restrict__ src, const int* __restrict__ dst,
    const float* __restrict__ el, const float* __restrict__ er,
    const unsigned* __restrict__ menc, const float* __restrict__ s,
    const float* __restrict__ feat, float* __restrict__ out, int E)
{
    int e    = (blockIdx.x * 256 + threadIdx.x) >> 5;
    int lane = threadIdx.x & 31;
    if (e >= E) return;
    int sn = src[e], dn = dst[e];
    float v     = leaky(el[sn] + er[dn]);
    float alpha = expf(v - dec_ord(menc[dn])) / s[dn];
    float4 f = ((const float4*)(feat + (size_t)sn * D_H))[lane];
    float* o = out + (size_t)dn * D_H + lane * 4;
    atomicAdd(o + 0, alpha * f.x);
    atomicAdd(o + 1, alpha * f.y);
    atomicAdd(o + 2, alpha * f.z);
    atomicAdd(o + 3, alpha * f.w);
}

// ---------------------------------------------------------------------------
// x = relu(in + bias), elementwise over N*128 (nvec4 = N*32 float4s)
// ---------------------------------------------------------------------------
__global__ void __launch_bounds__(256) bias_relu(
    const float* __restrict__ in, const float* __restrict__ b,
    float* __restrict__ out, int nvec4)
{
    int i = blockIdx.x * 256 + threadIdx.x;
    if (i >= nvec4) return;
    int f4 = i & 31;
    float4 v  = ((const float4*)in)[i];
    float4 bb = ((const float4*)b)[f4];
    v.x = fmaxf(v.x + bb.x, 0.f);
    v.y = fmaxf(v.y + bb.y, 0.f);
    v.z = fmaxf(v.z + bb.z, 0.f);
    v.w = fmaxf(v.w + bb.w, 0.f);
    ((float4*)out)[i] = v;
}

// ---------------------------------------------------------------------------
// per-graph sum + count via LDS pre-reduction (graph_id sorted -> few graphs/block)
// ---------------------------------------------------------------------------
__global__ void __launch_bounds__(256) pool_kernel(
    const float* __restrict__ x, const int* __restrict__ gid,
    float* __restrict__ hg, float* __restrict__ cnt, int N)
{
    __shared__ float lhg[NUM_GRAPHS * D_H];
    __shared__ float lcnt[NUM_GRAPHS];
    for (int i = threadIdx.x; i < NUM_GRAPHS * D_H; i += 256) lhg[i] = 0.f;
    if (threadIdx.x < NUM_GRAPHS) lcnt[threadIdx.x] = 0.f;
    __syncthreads();

    int node0 = blockIdx.x * 256;
    int f   = threadIdx.x & 127;
    int sub = threadIdx.x >> 7;
    for (int i = 0; i < 128; ++i) {
        int n = node0 + i * 2 + sub;
        if (n < N) {
            int g = gid[n];
            atomicAdd(&lhg[g * D_H + f], x[(size_t)n * D_H + f]);
            if (f == 0) atomicAdd(&lcnt[g], 1.f);
        }
    }
    __syncthreads();

    for (int i = threadIdx.x; i < NUM_GRAPHS * D_H; i += 256) {
        float v = lhg[i];
        if (v != 0.f) atomicAdd(&hg[i], v);
    }
    if (threadIdx.x < NUM_GRAPHS) {
        float v = lcnt[threadIdx.x];
        if (v != 0.f) atomicAdd(&cnt[threadIdx.x], v);
    }
}

// ---------------------------------------------------------------------------
// mean -> logits (128x2) -> log_softmax. One thread per graph.
// ---------------------------------------------------------------------------
__global__ void head_kernel(
    const float* __restrict__ hg, const float* __restrict__ cnt,
    const float* __restrict__ Wfc, const float* __restrict__ bfc,
    float* __restrict__ out)
{
    int g = threadIdx.x;
    if (g >= NUM_GRAPHS) return;
    float c = cnt[g]; if (c < 1.f) c = 1.f;
    float z0 = bfc[0], z1 = bfc[1];
    for (int f = 0; f < D_H; ++f) {
        float v = hg[g * D_H + f] / c;
        z0 = fmaf(v, Wfc[f * 2 + 0], z0);
        z1 = fmaf(v, Wfc[f * 2 + 1], z1);
    }
    float mx  = fmaxf(z0, z1);
    float lse = mx + logf(expf(z0 - mx) + expf(z1 - mx));
    out[g * 2 + 0] = z0 - lse;
    out[g * 2 + 1] = z1 - lse;
}

// ---------------------------------------------------------------------------
extern "C" void kernel_launch(void* const* d_in, const int* in_sizes, int n_in,
                              void* d_out, int out_size, void* d_ws, size_t ws_size,
                              hipStream_t stream)
{
    const int N = in_sizes[0] / D_H;
    const int E = in_sizes[1];

    const float* h   = (const float*)d_in[0];
    const int*   src = (const int*)  d_in[1];
    const int*   dst = (const int*)  d_in[2];
    const int*   gid = (const int*)  d_in[3];
    const float* W1  = (const float*)d_in[4];
    const float* al1 = (const float*)d_in[5];
    const float* ar1 = (const float*)d_in[6];
    const float* b1  = (const float*)d_in[7];
    const float* W2  = (const float*)d_in[8];
    const float* al2 = (const float*)d_in[9];
    const float* ar2 = (const float*)d_in[10];
    const float* b2  = (const float*)d_in[11];
    const float* Wfc = (const float*)d_in[12];
    const float* bfc = (const float*)d_in[13];

    const size_t nd = (size_t)N * D_H;
    float*    A    = (float*)d_ws;              // feat1 / x2 / out2
    float*    B    = A + nd;                    // out1 / feat2
    float*    el   = B + nd;
    float*    er   = el + N;
    unsigned* menc = (unsigned*)(er + N);
    float*    s    = (float*)(menc + N);
    float*    hg   = s + N;
    float*    cnt  = hg + NUM_GRAPHS * D_H;

    dim3 blk(256);
    const int gemmBlocks     = (N + 15) / 16;
    const int nodeWaveBlocks = (int)(((size_t)N * 32 + 255) / 256);
    const int edgeBlocks     = (E + 255) / 256;
    const int edgeWaveBlocks = (int)(((size_t)E * 32 + 255) / 256);
    const int nvec4          = N * 32;
    const int reluBlocks     = (nvec4 + 255) / 256;
    const int poolBlocks     = (N + 255) / 256;

    // ---------------- layer 1 ----------------
    hipMemsetAsync(B,    0, nd * sizeof(float), stream);         // out1 accumulator
    hipMemsetAsync(menc, 0, (size_t)N * 4, stream);              // enc(-inf): 0 loses to every real value
    hipMemsetAsync(s,    0, (size_t)N * 4, stream);
    gat_gemm_wmma<<<gemmBlocks, blk, 0, stream>>>(h, W1, A, N);  // feat1 = A
    node_attn<<<nodeWaveBlocks, blk, 0, stream>>>(A, al1, ar1, el, er, N);
    edge_max<<<edgeBlocks, blk, 0, stream>>>(src, dst, el, er, menc, E);
    edge_sum<<<edgeBlocks, blk, 0, stream>>>(src, dst, el, er, menc, s, E);
    edge_aggr<<<edgeWaveBlocks, blk, 0, stream>>>(src, dst, el, er, menc, s, A, B, E);
    bias_relu<<<reluBlocks, blk, 0, stream>>>(B, b1, A, nvec4);  // x2 = relu(out1+b1) -> A

    // ---------------- layer 2 ----------------
    hipMemsetAsync(menc, 0, (size_t)N * 4, stream);
    hipMemsetAsync(s,    0, (size_t)N * 4, stream);
    gat_gemm_wmma<<<gemmBlocks, blk, 0, stream>>>(A, W2, B, N);  // feat2 = B
    hipMemsetAsync(A, 0, nd * sizeof(float), stream);            // out2 accumulator = A
    node_attn<<<nodeWaveBlocks, blk, 0, stream>>>(B, al2, ar2, el, er, N);
    edge_max<<<edgeBlocks, blk, 0, stream>>>(src, dst, el, er, menc, E);
    edge_sum<<<edgeBlocks, blk, 0, stream>>>(src, dst, el, er, menc, s, E);
    edge_aggr<<<edgeWaveBlocks, blk, 0, stream>>>(src, dst, el, er, menc, s, B, A, E);
    bias_relu<<<reluBlocks, blk, 0, stream>>>(A, b2, A, nvec4);  // x_out = relu(out2+b2) -> A

    // ---------------- readout ----------------
    hipMemsetAsync(hg,  0, NUM_GRAPHS * D_H * sizeof(float), stream);
    hipMemsetAsync(cnt, 0, NUM_GRAPHS * sizeof(float), stream);
    pool_kernel<<<poolBlocks, blk, 0, stream>>>(A, gid, hg, cnt, N);
    head_kernel<<<1, 64, 0, stream>>>(hg, cnt, Wfc, bfc, (float*)d_out);
}